// SuperpointGenerator_48679159333096
// MI455X (gfx1250) — compile-verified
//
#include <hip/hip_runtime.h>
#include <stdint.h>

#define NB      16
#define NPTS    262144
#define KTOP    256
#define DLOG    21
#define DSIZE   (1u << DLOG)        // 2,097,152 bins
#define VOFF    (1 << 20)
#define CAP     2047                // 256*CAP > NPTS -> threshold always < CAP
#define CCH     2048                // ccHist size (index 1..CAP used)
#define NCHUNK  2048                // DSIZE / 1024
#define AUXN    (5*2048 + 16 + 512) // cc,G,chunkOcc,chunkPfx,chunkT,scal,selKeys
#define NZ4     (DSIZE / 4)

typedef unsigned int u32x4 __attribute__((ext_vector_type(4)));
typedef int          i32x4 __attribute__((ext_vector_type(4)));
typedef int          i32x8 __attribute__((ext_vector_type(8)));

// ---------------- block-wide inclusive scan over 1024 lanes (LDS) -----------
__device__ __forceinline__ int blockIncScan1024(int val, int* sbuf, int t) {
    sbuf[t] = val;
    __syncthreads();
    #pragma unroll
    for (int off = 1; off < 1024; off <<= 1) {
        int y = (t >= off) ? sbuf[t - off] : 0;
        __syncthreads();
        sbuf[t] += y;
        __syncthreads();
    }
    return sbuf[t];
}

// ---------------- zero histogram (b128 stores) + aux ------------------------
__global__ void kZero(int4* __restrict__ hist4, int* __restrict__ aux) {
    unsigned i = blockIdx.x * 1024u + threadIdx.x;
    if (i < NZ4) hist4[i] = make_int4(0, 0, 0, 0);
    else { unsigned j = i - NZ4; if (j < AUXN) aux[j] = 0; }
}

// ---------------- hash points, build bin histogram --------------------------
// Stages each block's 1024 points (12KB) into LDS with ONE Tensor-Data-Mover
// DMA (wave 0 builds the D# descriptor), falling back to per-lane async
// global->LDS copies when the TDM builtin is unavailable.
__global__ void kHash(const float* __restrict__ coords,
                      int* __restrict__ hist, int* __restrict__ ids) {
    __shared__ float tile[3072];
    const int t = threadIdx.x;
    const unsigned p = blockIdx.x * 1024u + t;
#if __has_builtin(__builtin_amdgcn_tensor_load_to_lds)
    if (t < 32) {                            // wave 0 issues the DMA
        unsigned ldsb = (unsigned)(size_t)(&tile[0]);     // LDS byte offset
        unsigned long long ga =
            (unsigned long long)(size_t)(coords + (size_t)blockIdx.x * 3072u);
        u32x4 g0 = { 1u,                                  // count=1 user D#
                     ldsb,                                // lds_addr
                     (unsigned)(ga & 0xFFFFFFFFull),      // global_addr lo
                     (unsigned)((ga >> 32) & 0x01FFFFFFull) | 0x80000000u }; // type=2
        i32x8 g1 = { 0x00020000,             // data_size=4B; mask/pad/iter=0
                     (int)(3072u << 16),     // tensor_dim0[15:0] in [31:16]
                     (int)(1u << 16),        // tensor_dim0 hi=0 | tensor_dim1=1
                     (int)(3072u << 16),     // tensor_dim1 hi=0 | tile_dim0=3072
                     1,                      // tile_dim1=1 | tile_dim2=0
                     3072,                   // tensor_dim0_stride lo
                     0, 0 };                 // stride0 hi | stride1
        i32x4 gz = { 0, 0, 0, 0 };
#if __clang_major__ >= 23
        i32x8 gz8 = { 0, 0, 0, 0, 0, 0, 0, 0 };
        __builtin_amdgcn_tensor_load_to_lds(g0, g1, gz, gz, gz8, 0);
#else
        __builtin_amdgcn_tensor_load_to_lds(g0, g1, gz, gz, 0);
#endif
#if __has_builtin(__builtin_amdgcn_s_wait_tensorcnt)
        __builtin_amdgcn_s_wait_tensorcnt(0);
#else
        asm volatile("s_wait_tensorcnt 0" ::: "memory");
#endif
    }
#else
    {   // fallback: per-lane async global->LDS copies
        const float* src = coords + (size_t)p * 3;
        unsigned lb = (unsigned)(size_t)(&tile[t * 3]);
        asm volatile("global_load_async_to_lds_b32 %0, %1, off"
                     :: "v"(lb),      "v"(src)     : "memory");
        asm volatile("global_load_async_to_lds_b32 %0, %1, off"
                     :: "v"(lb + 4u), "v"(src + 1) : "memory");
        asm volatile("global_load_async_to_lds_b32 %0, %1, off"
                     :: "v"(lb + 8u), "v"(src + 2) : "memory");
        asm volatile("s_wait_asynccnt 0" ::: "memory");
    }
#endif
    __syncthreads();
    float x = tile[t * 3 + 0], y = tile[t * 3 + 1], z = tile[t * 3 + 2];
    int fx = (int)floorf(x / 0.2f);
    int fy = (int)floorf(y / 0.2f);
    int fz = (int)floorf(z / 0.2f);
    int v = fx * 10000 + fy * 100 + fz + VOFF;
    v = v < 0 ? 0 : (v >= (int)DSIZE ? (int)DSIZE - 1 : v);  // defensive clamp
    ids[p] = v;
    atomicAdd(&hist[v], 1);
}

// ---------------- per-chunk occupancy + count-of-counts (LDS privatized) ----
__global__ void kStats(const int* __restrict__ hist, int* __restrict__ ccHist,
                       int* __restrict__ chunkOcc) {
    __shared__ int lcc[CCH];
    __shared__ int occCnt;
    const int t = threadIdx.x;
    for (int i = t; i < CCH; i += 1024) lcc[i] = 0;
    if (t == 0) occCnt = 0;
    __syncthreads();
    unsigned v = blockIdx.x * 1024u + t;
    __builtin_prefetch(&hist[v + 65536u], 0, 0);           // global_prefetch_b8
    int c = hist[v];
    if (c > 0) atomicAdd(&lcc[c < CAP ? c : CAP], 1);      // ds_add_u32
    unsigned mask = (unsigned)__ballot(c > 0);             // wave32 ballot
    if ((t & 31) == 0) atomicAdd(&occCnt, __popc(mask));
    __syncthreads();
    for (int i = t; i < CCH; i += 1024)
        if (lcc[i]) atomicAdd(&ccHist[i], lcc[i]);
    if (t == 0) chunkOcc[blockIdx.x] = occCnt;
}

// ---------------- one-block parallel: chunk prefix, suffix sums, T, nT ------
__global__ void kScanA(const int* __restrict__ chunkOcc, int* __restrict__ chunkPfx,
                       const int* __restrict__ ccHist, int* __restrict__ G,
                       int* __restrict__ scal) {
    __shared__ int sb[1024];
    __shared__ int sT;
    const int t = threadIdx.x;
    // exclusive scan of chunkOcc[2048], 2 per thread
    int a = chunkOcc[2 * t], b = chunkOcc[2 * t + 1];
    int incl = blockIncScan1024(a + b, sb, t);
    int excl = incl - (a + b);
    chunkPfx[2 * t]     = excl;
    chunkPfx[2 * t + 1] = excl + a;
    __syncthreads();
    int U = sb[1023];
    __syncthreads();
    // suffix sums G[c] = sum_{c'>=c} cc[c'] via reversed forward scan
    int c0 = CAP - 2 * t, c1 = CAP - (2 * t + 1);
    int v0 = (c0 >= 1) ? ccHist[c0] : 0;
    int v1 = (c1 >= 1) ? ccHist[c1] : 0;
    int incl2 = blockIncScan1024(v0 + v1, sb, t);
    int e2  = incl2 - (v0 + v1);
    int g0v = e2 + v0;                 // G[c0]
    int g1v = e2 + v0 + v1;            // G[c1]
    if (c0 >= 1) G[c0] = g0v;
    if (c1 >= 1) G[c1] = g1v;
    if (t == 0) sT = 1;
    __syncthreads();
    if (c0 >= 1 && g0v >= KTOP) atomicMax(&sT, c0);   // ds_max
    if (c1 >= 1 && g1v >= KTOP) atomicMax(&sT, c1);
    __syncthreads();
    if (t == 0) {
        int T = sT;
        scal[0] = U;
        scal[5] = (U <= KTOP) ? 1 : 0;
        scal[1] = T;
        int GT1 = (T < CAP) ? G[T + 1] : 0;    // #bins with cnt > T
        scal[2] = KTOP - GT1;                  // nT
        scal[4] = 0;                           // selCount
    }
}

// ---------------- per-chunk count of bins with cnt == T ---------------------
__global__ void kChunkT(const int* __restrict__ hist, const int* __restrict__ scal,
                        int* __restrict__ chunkT) {
    __shared__ int cnt;
    if (threadIdx.x == 0) cnt = 0;
    __syncthreads();
    int T = scal[1];
    unsigned v = blockIdx.x * 1024u + threadIdx.x;
    unsigned mask = (unsigned)__ballot(hist[v] == T);
    if ((threadIdx.x & 31) == 0) atomicAdd(&cnt, __popc(mask));
    __syncthreads();
    if (threadIdx.x == 0) chunkT[blockIdx.x] = cnt;
}

// ---------------- one-block parallel: locate v-cutoff among count==T bins ---
__global__ void kScanB(const int* __restrict__ hist, const int* __restrict__ chunkT,
                       int* __restrict__ scal) {
    __shared__ int sb[1024];
    __shared__ int sj, sbase;
    const int t = threadIdx.x;
    int useInv = scal[5];
    int nT = scal[2], T = scal[1];
    if (t == 0) { sj = 0; sbase = 0; scal[3] = -1; scal[4] = 0; }
    __syncthreads();
    if (useInv || nT <= 0) return;          // uniform exit
    int a = chunkT[2 * t], b = chunkT[2 * t + 1];
    int incl = blockIncScan1024(a + b, sb, t);
    int excl = incl - (a + b);
    if (excl < nT && nT <= excl + a)             { sj = 2 * t;     sbase = excl; }
    if (excl + a < nT && nT <= excl + a + b)     { sj = 2 * t + 1; sbase = excl + a; }
    __syncthreads();
    int j = sj, base = sbase;
    unsigned v = (unsigned)j * 1024u + t;
    int is = (hist[v] == T) ? 1 : 0;
    int incl2 = blockIncScan1024(is, sb, t);     // rank within chunk (1-based)
    if (is && (base + incl2 == nT)) scal[3] = (int)v;
}

// ---------------- fill label map (dense ranks or -1) + gather winners -------
__global__ void kFill(const int* __restrict__ hist, int* __restrict__ labelMap,
                      const int* __restrict__ chunkPfx, int* __restrict__ scal,
                      unsigned long long* __restrict__ selKeys) {
    __shared__ int wsum[32];
    const int t = threadIdx.x;
    unsigned v = blockIdx.x * 1024u + t;
    int c = hist[v];
    int occ = (c > 0);
    if (scal[5]) {                       // U <= K: labels are dense ranks
        unsigned mask = (unsigned)__ballot(occ);
        int lane = t & 31, wid = t >> 5;
        if (lane == 0) wsum[wid] = __popc(mask);
        __syncthreads();
        if (t == 0) { int r = 0; for (int w = 0; w < 32; ++w) { int x = wsum[w]; wsum[w] = r; r += x; } }
        __syncthreads();
        int rank = chunkPfx[blockIdx.x] + wsum[wid] + __popc(mask & ((1u << lane) - 1u));
        labelMap[v] = occ ? rank : -1;
    } else {
        labelMap[v] = -1;
        int T = scal[1], vcut = scal[3];
        if (occ && (c > T || (c == T && (int)v <= vcut))) {
            int pos = atomicAdd(&scal[4], 1);
            if (pos < KTOP)              // key: (cnt desc, v asc) -> ascending
                selKeys[pos] = ((unsigned long long)(0x100000 - c) << 21) | (unsigned long long)v;
        }
    }
}

// ---------------- sort 256 winners, scatter labels --------------------------
__global__ void kSort(int* __restrict__ labelMap, const int* __restrict__ scal,
                      const unsigned long long* __restrict__ selKeys) {
    __shared__ unsigned long long sk[KTOP];
    if (scal[5]) return;                 // uniform exit
    const int t = threadIdx.x;
    int n = scal[4]; if (n > KTOP) n = KTOP;
    sk[t] = (t < n) ? selKeys[t] : ~0ULL;
    __syncthreads();
    for (int k = 2; k <= KTOP; k <<= 1)
        for (int j = k >> 1; j > 0; j >>= 1) {
            int ixj = t ^ j;
            if (ixj > t) {
                unsigned long long a = sk[t], b = sk[ixj];
                if ((a > b) == ((t & k) == 0)) { sk[t] = b; sk[ixj] = a; }
            }
            __syncthreads();
        }
    if (t < n) labelMap[(unsigned)(sk[t] & 0x1FFFFFu)] = t;
}

// ---------------- final gather: out[i] = labelMap[ids[i]] -------------------
// ids staged through LDS with per-lane CDNA5 async global->LDS copies.
__global__ void kOut(const int* __restrict__ ids, const int* __restrict__ labelMap,
                     int* __restrict__ out) {
    __shared__ int sid[1024];
    const int t = threadIdx.x;
    unsigned i = blockIdx.x * 1024u + t;
    unsigned lb = (unsigned)(size_t)(&sid[t]);
    const int* src = &ids[i];
    asm volatile("global_load_async_to_lds_b32 %0, %1, off"
                 :: "v"(lb), "v"(src) : "memory");
    asm volatile("s_wait_asynccnt 0" ::: "memory");
    out[i] = labelMap[sid[t]];
}

extern "C" void kernel_launch(void* const* d_in, const int* in_sizes, int n_in,
                              void* d_out, int out_size, void* d_ws, size_t ws_size,
                              hipStream_t stream) {
    (void)in_sizes; (void)n_in; (void)out_size; (void)ws_size;
    const float* coords = (const float*)d_in[0];
    int* out = (int*)d_out;

    int* ws       = (int*)d_ws;
    int* hist     = ws;                           // DSIZE ints
    int* labelMap = ws + DSIZE;                   // DSIZE ints
    int* ids      = ws + 2 * DSIZE;               // NPTS ints
    int* aux      = ws + 2 * DSIZE + NPTS;        // AUXN ints
    int* ccHist   = aux;                          // 2048
    int* G        = aux + 2048;                   // 2048
    int* chunkOcc = aux + 2 * 2048;               // 2048
    int* chunkPfx = aux + 3 * 2048;               // 2048
    int* chunkT   = aux + 4 * 2048;               // 2048
    int* scal     = aux + 5 * 2048;               // 16
    unsigned long long* selKeys = (unsigned long long*)(scal + 16);  // 256 u64

    const int zgrid = (int)((NZ4 + AUXN + 1023) / 1024);
    for (int b = 0; b < NB; ++b) {
        kZero  <<<zgrid,       1024, 0, stream>>>((int4*)hist, aux);
        kHash  <<<NPTS / 1024, 1024, 0, stream>>>(coords + (size_t)b * NPTS * 3, hist, ids);
        kStats <<<NCHUNK,      1024, 0, stream>>>(hist, ccHist, chunkOcc);
        kScanA <<<1,           1024, 0, stream>>>(chunkOcc, chunkPfx, ccHist, G, scal);
        kChunkT<<<NCHUNK,      1024, 0, stream>>>(hist, scal, chunkT);
        kScanB <<<1,           1024, 0, stream>>>(hist, chunkT, scal);
        kFill  <<<NCHUNK,      1024, 0, stream>>>(hist, labelMap, chunkPfx, scal, selKeys);
        kSort  <<<1,           KTOP, 0, stream>>>(labelMap, scal, selKeys);
        kOut   <<<NPTS / 1024, 1024, 0, stream>>>(ids, labelMap, out + (size_t)b * NPTS);
    }
}